// Hausdorff_Distance_Binary_Image_35416300323044
// MI455X (gfx1250) — compile-verified
//
#include <hip/hip_runtime.h>
#include <hip/hip_bf16.h>
#include <math.h>

// ---------------------------------------------------------------------------
// Directed Hausdorff distance on binary boundary images, N=4, C=2, H=W=96.
// Core: d2(a,b) = |a|^2 + |b|^2 - 2 a.b  ==> rank-4 GEMM per 16x16 tile via
// V_WMMA_F32_16X16X4_F32 (exact in f32 for coords <= 95).
//   WMMA A-operand rows  = label-boundary points  q = ( bx,  by, |b|^2, 1    )
//   WMMA B-operand cols  = pred-boundary points   p = (-2ax,-2ay, 1,    |a|^2)
//   D[j][i] = q_j . p_i = squared distance (exact).
// Row pads: (0,0,+BIG,1)  -> d2 ~ +BIG, never wins the min.
// Col pads: (0,0,0,-BIG)  -> d2 = -BIG, never wins the max.
// ---------------------------------------------------------------------------

typedef __attribute__((ext_vector_type(2))) float v2f;
typedef __attribute__((ext_vector_type(8))) float v8f;

#define Hn   96
#define Wn   96
#define HWn  (Hn * Wn)       // 9216
#define Nn   4
#define CAP  9232            // 577 * 16 : max points (9216) rounded up + pad room
#define BIGF 1.0e30f

// workspace layout (bytes):
//   0   : int   cntA[4]   (real a-point counts)
//   16  : int   cntB[4]   (real b-point counts)
//   32  : int   padA[4]   (padded-to-16 a counts)
//   48  : int   padB[4]   (padded-to-16 b counts)
//   64  : float hd2[4]    (atomic-max accumulators, bit-pattern int max)
//   256 : float P[4][CAP*4]   a-point (column) operands
//   256 + 4*CAP*16 : float Q[4][CAP*4]   b-point (row) operands
#define P_BASE 256
#define Q_BASE (256 + (size_t)Nn * CAP * 16)

// ---------------------------------------------------------------------------
// Kernel 1: build boundary masks, compact points into WMMA-ready 4-float
// records, pad each list to a multiple of 16, zero the hd2 accumulators.
// One block per example; masks staged in LDS (2 x 9216 B << 320 KB/WGP).
// ---------------------------------------------------------------------------
__global__ void hd_build_points(const float* __restrict__ pred,
                                const float* __restrict__ lab,
                                char* __restrict__ ws) {
    const int n = blockIdx.x;
    __shared__ unsigned char am[HWn];
    __shared__ unsigned char bm[HWn];
    __shared__ int cA, cB;
    if (threadIdx.x == 0) { cA = 0; cB = 0; }

    const float* p0 = pred + ((size_t)n * 2 + 0) * HWn;  // class-0 logits
    const float* p1 = pred + ((size_t)n * 2 + 1) * HWn;  // class-1 logits
    const float* l1 = lab  + ((size_t)n * 2 + 1) * HWn;  // class-1 labels (0/1)

    for (int p = threadIdx.x; p < HWn; p += blockDim.x) {
        am[p] = (unsigned char)(p1[p] > p0[p]);   // argmax==1 (ties -> class 0)
        bm[p] = (unsigned char)(l1[p] > 0.5f);
    }
    __syncthreads();

    int*   cntA = (int*)(ws);
    int*   cntB = (int*)(ws + 16);
    int*   padA = (int*)(ws + 32);
    int*   padB = (int*)(ws + 48);
    int*   hd2  = (int*)(ws + 64);
    float* P    = (float*)(ws + P_BASE) + (size_t)n * CAP * 4;
    float* Q    = (float*)(ws + Q_BASE) + (size_t)n * CAP * 4;

    for (int p = threadIdx.x; p < HWn; p += blockDim.x) {
        const int h = p / Wn, w = p % Wn;
        // interior <=> all 4 neighbors exist and are set (off-grid counts 0)
        const bool intA = (h > 0 && am[p - Wn]) && (h < Hn - 1 && am[p + Wn]) &&
                          (w > 0 && am[p - 1])  && (w < Wn - 1 && am[p + 1]);
        const bool intB = (h > 0 && bm[p - Wn]) && (h < Hn - 1 && bm[p + Wn]) &&
                          (w > 0 && bm[p - 1])  && (w < Wn - 1 && bm[p + 1]);
        const bool aB = am[p] && !intA && !bm[p];   // * (1 - b_img)
        const bool bB = bm[p] && !intB;
        const float fh = (float)h, fw = (float)w;
        if (aB) {
            const int idx = atomicAdd(&cA, 1);
            float* d = P + (size_t)idx * 4;
            d[0] = -2.0f * fh; d[1] = -2.0f * fw; d[2] = 1.0f; d[3] = fh * fh + fw * fw;
        }
        if (bB) {
            const int idx = atomicAdd(&cB, 1);
            float* d = Q + (size_t)idx * 4;
            d[0] = fh; d[1] = fw; d[2] = fh * fh + fw * fw; d[3] = 1.0f;
        }
    }
    __syncthreads();

    if (threadIdx.x == 0) {
        const int a = cA, b = cB;
        const int ap = (a + 15) & ~15, bp = (b + 15) & ~15;
        for (int i = a; i < ap; ++i) {           // col pads: d2 = -BIG
            float* d = P + (size_t)i * 4;
            d[0] = 0.0f; d[1] = 0.0f; d[2] = 0.0f; d[3] = -BIGF;
        }
        for (int i = b; i < bp; ++i) {           // row pads: d2 = +BIG
            float* d = Q + (size_t)i * 4;
            d[0] = 0.0f; d[1] = 0.0f; d[2] = BIGF; d[3] = 1.0f;
        }
        cntA[n] = a; cntB[n] = b; padA[n] = ap; padB[n] = bp;
        hd2[n] = 0;  // 0 bits == 0.0f
    }
}

// ---------------------------------------------------------------------------
// Kernel 2: each wave owns one 16-column tile of A-points, streams all
// B-point row tiles through V_WMMA_F32_16X16X4_F32, keeps a per-lane running
// min (each lane holds one column, 8 rows per tile in the D VGPRs), then
// merges halves / maxes across columns and atomic-maxes into hd2[n].
// ---------------------------------------------------------------------------
__global__ void hd_wmma_minmax(char* __restrict__ ws) {
    const int n    = blockIdx.y;
    const int wave = threadIdx.x >> 5;
    const int lane = threadIdx.x & 31;
    const int tile = blockIdx.x * (blockDim.x >> 5) + wave;

    const int* padA = (const int*)(ws + 32);
    const int* padB = (const int*)(ws + 48);
    int*       hd2  = (int*)(ws + 64);
    const int cApad = padA[n];
    const int cBpad = padB[n];
    const int i0    = tile * 16;
    if (i0 >= cApad || cBpad == 0) return;       // wave-uniform exit

    const float4* P = (const float4*)(ws + P_BASE) + (size_t)n * CAP;
    const float4* Q = (const float4*)(ws + Q_BASE) + (size_t)n * CAP;

    const int  L  = lane & 15;
    const bool hi = lane >= 16;                  // lanes 16..31 carry K=2,3

    // B operand (columns = a-points), resident for the whole loop
    const float4 pv = P[i0 + L];
    v2f bop;
    bop.x = hi ? pv.z : pv.x;
    bop.y = hi ? pv.w : pv.y;

    float rmin = BIGF;
    for (int j0 = 0; j0 < cBpad; j0 += 16) {
        const float4 qv = Q[j0 + L];             // A operand rows = b-points
        v2f aop;
        aop.x = hi ? qv.z : qv.x;
        aop.y = hi ? qv.w : qv.y;

        v8f c = {};
        c = __builtin_amdgcn_wmma_f32_16x16x4_f32(
                /*neg_a=*/false, aop, /*neg_b=*/false, bop,
                /*c_mod=*/(short)0, c, /*reuse_a=*/false, /*reuse_b=*/false);

        // min over the 8 rows this lane holds for its column
        const float m = fminf(fminf(fminf(c[0], c[1]), fminf(c[2], c[3])),
                              fminf(fminf(c[4], c[5]), fminf(c[6], c[7])));
        rmin = fminf(rmin, m);
    }

    // lanes L and L+16 hold disjoint row halves of the same column -> min
    rmin = fminf(rmin, __shfl_xor(rmin, 16, 32));
    // max across the 16 columns (both halves now identical)
    float mx = rmin;
    mx = fmaxf(mx, __shfl_xor(mx, 8, 32));
    mx = fmaxf(mx, __shfl_xor(mx, 4, 32));
    mx = fmaxf(mx, __shfl_xor(mx, 2, 32));
    mx = fmaxf(mx, __shfl_xor(mx, 1, 32));

    if (lane == 0) {
        // values are >= 0 (a-points and b-points are disjoint pixel sets),
        // so int bit-pattern max == float max
        atomicMax(&hd2[n], __float_as_int(fmaxf(mx, 0.0f)));
    }
}

// ---------------------------------------------------------------------------
// Kernel 3: assemble the [N, C+2] output.
// col0 = 0 (ignored class), col1 = hd, col2 = mean over C = hd/2,
// col3 = mean over first C-1 columns = 0.
// ---------------------------------------------------------------------------
__global__ void hd_finalize(const char* __restrict__ ws, float* __restrict__ out) {
    const int n = threadIdx.x;
    if (n >= Nn) return;
    const int*   cntA = (const int*)(ws);
    const int*   cntB = (const int*)(ws + 16);
    const float* hd2  = (const float*)(ws + 64);
    const bool valid = (cntA[n] > 0) && (cntB[n] > 0);
    const float hd = valid ? sqrtf(fmaxf(hd2[n], 0.0f))
                           : (float)((Hn + Wn) / 4.0);   // 48.0 fallback
    out[n * 4 + 0] = 0.0f;
    out[n * 4 + 1] = hd;
    out[n * 4 + 2] = 0.5f * hd;
    out[n * 4 + 3] = 0.0f;
}

extern "C" void kernel_launch(void* const* d_in, const int* in_sizes, int n_in,
                              void* d_out, int out_size, void* d_ws, size_t ws_size,
                              hipStream_t stream) {
    const float* pred = (const float*)d_in[0];   // [4,2,96,96] f32
    const float* lab  = (const float*)d_in[1];   // [4,2,96,96] f32
    char*  ws  = (char*)d_ws;
    float* out = (float*)d_out;                  // [4,4] f32

    hd_build_points<<<dim3(Nn), dim3(512), 0, stream>>>(pred, lab, ws);

    // 577 column tiles max per example, 8 waves (256 threads) per block
    dim3 grid((577 + 7) / 8, Nn);
    hd_wmma_minmax<<<grid, dim3(256), 0, stream>>>(ws);

    hd_finalize<<<dim3(1), dim3(32), 0, stream>>>(ws, out);
}